// UUSAGE_15247133901327
// MI455X (gfx1250) — compile-verified
//
#include <hip/hip_runtime.h>

typedef float v2f __attribute__((ext_vector_type(2)));
typedef float v8f __attribute__((ext_vector_type(8)));

#define NN 50000
#define NE 800000

// ---------------- zero fill ----------------
__global__ void zero_f32(float* __restrict__ p, int n) {
    int i = blockIdx.x * blockDim.x + threadIdx.x;
    if (i < n) p[i] = 0.0f;
}

// ---------------- degree (segment_sum of ones over dst) ----------------
__global__ void degree_kernel(const int* __restrict__ dst, float* __restrict__ cnt, int E) {
    int e = blockIdx.x * blockDim.x + threadIdx.x;
    if (e < E) atomicAdd(&cnt[dst[e]], 1.0f);
}

// ---------------- scatter-add of gathered neighbor rows ----------------
// one thread = one edge x 4 consecutive features; d = (1<<d_log4)*4
// consecutive 32 threads = one edge => coalesced 512B row gather, broadcast idx loads
__global__ void scatter_add_kernel(const float* __restrict__ h,
                                   const int* __restrict__ src,
                                   const int* __restrict__ dst,
                                   float* __restrict__ agg,
                                   int E, int d_log4) {
    int t = blockIdx.x * blockDim.x + threadIdx.x;
    if (t >= (E << d_log4)) return;
    int nchunk = 1 << d_log4;
    int e = t >> d_log4;
    int c = t & (nchunk - 1);
    int d = nchunk << 2;
    int s = src[e], q = dst[e];
    const float4 v = *(const float4*)(h + (size_t)s * d + (c << 2));
    float* p = agg + (size_t)q * d + (c << 2);
    atomicAdd(p + 0, v.x);
    atomicAdd(p + 1, v.y);
    atomicAdd(p + 2, v.z);
    atomicAdd(p + 3, v.w);
}

// ---------------- fused SAGE layer GEMM: out = (agg/cnt)@Wl + x@Wr + b ----------------
// One wave32 per PAIR of 16x16 output tiles stacked in M, sharing B fragments.
// V_WMMA_F32_16X16X4_F32, K-step = 4.
// A layout (16x4 f32): lanes 0-15 hold K={k0,k0+1}, lanes 16-31 hold K={k0+2,k0+3}, M=lane&15.
// B layout (4x16 f32): mirrored, N=lane&15.
// C/D (16x16 f32): vgpr r -> row = 16*tm + 8*(lane>>4) + r, col = 16*tn + (lane&15).
__global__ void sage_gemm_wmma(const float* __restrict__ agg,
                               const float* __restrict__ cnt,
                               const float* __restrict__ xin,
                               const float* __restrict__ Wl,
                               const float* __restrict__ Wr,
                               const float* __restrict__ bias,
                               float* __restrict__ out,
                               int M, int K, int N, int relu) {
    const int wave = (blockIdx.x * blockDim.x + threadIdx.x) >> 5;
    const int lane = threadIdx.x & 31;
    const int tilesN = N >> 4;
    const int tilesM = M >> 4;
    const int pairsM = (tilesM + 1) >> 1;
    if (wave >= pairsM * tilesN) return;   // wave-uniform: EXEC stays all-1s for WMMA

    const int pm = wave / tilesN;
    const int tn = wave - pm * tilesN;
    const int tm0 = pm * 2;
    const int tm1raw = tm0 + 1;
    const bool has2 = (tm1raw < tilesM);          // wave-uniform
    const int tm1 = has2 ? tm1raw : (tilesM - 1); // clamp: tiny redundant work on last pair

    const int r15   = lane & 15;
    const int hi    = lane >> 4;
    const int row0  = tm0 * 16 + r15;
    const int row1  = tm1 * 16 + r15;
    const int colB  = tn * 16 + r15;
    const int khalf = hi * 2;

    const float inv0 = 1.0f / fmaxf(cnt[row0], 1.0f);
    const float inv1 = 1.0f / fmaxf(cnt[row1], 1.0f);
    const float* __restrict__ aggRow0 = agg + (size_t)row0 * K;
    const float* __restrict__ xRow0   = xin + (size_t)row0 * K;
    const float* __restrict__ aggRow1 = agg + (size_t)row1 * K;
    const float* __restrict__ xRow1   = xin + (size_t)row1 * K;

    v8f c0 = {0.f, 0.f, 0.f, 0.f, 0.f, 0.f, 0.f, 0.f};
    v8f d0 = {0.f, 0.f, 0.f, 0.f, 0.f, 0.f, 0.f, 0.f};
    v8f c1 = {0.f, 0.f, 0.f, 0.f, 0.f, 0.f, 0.f, 0.f};
    v8f d1 = {0.f, 0.f, 0.f, 0.f, 0.f, 0.f, 0.f, 0.f};

    for (int k0 = 0; k0 < K; k0 += 4) {
        const int kk = k0 + khalf;
        // shared B fragments (one set for both M-tiles)
        v2f bL, bR;
        bL.x = Wl[(size_t)kk * N + colB];
        bL.y = Wl[(size_t)(kk + 1) * N + colB];
        bR.x = Wr[(size_t)kk * N + colB];
        bR.y = Wr[(size_t)(kk + 1) * N + colB];
        // tile 0
        v2f aM0 = *(const v2f*)(aggRow0 + kk);
        aM0.x *= inv0; aM0.y *= inv0;             // fused mean
        v2f aX0 = *(const v2f*)(xRow0 + kk);
        // tile 1
        v2f aM1 = *(const v2f*)(aggRow1 + kk);
        aM1.x *= inv1; aM1.y *= inv1;
        v2f aX1 = *(const v2f*)(xRow1 + kk);

        c0 = __builtin_amdgcn_wmma_f32_16x16x4_f32(false, aM0, false, bL, (short)0, c0, false, false);
        d0 = __builtin_amdgcn_wmma_f32_16x16x4_f32(false, aX0, false, bR, (short)0, d0, false, false);
        c1 = __builtin_amdgcn_wmma_f32_16x16x4_f32(false, aM1, false, bL, (short)0, c1, false, false);
        d1 = __builtin_amdgcn_wmma_f32_16x16x4_f32(false, aX1, false, bR, (short)0, d1, false, false);
    }

    const float bn = bias[colB];
    {
        const int rbase = tm0 * 16 + hi * 8;
#pragma unroll
        for (int r = 0; r < 8; ++r) {
            float v = c0[r] + d0[r] + bn;
            if (relu) v = fmaxf(v, 0.0f);
            out[(size_t)(rbase + r) * N + colB] = v;
        }
    }
    if (has2) {
        const int rbase = tm1 * 16 + hi * 8;
#pragma unroll
        for (int r = 0; r < 8; ++r) {
            float v = c1[r] + d1[r] + bn;
            if (relu) v = fmaxf(v, 0.0f);
            out[(size_t)(rbase + r) * N + colB] = v;
        }
    }
}

// ---------------- log_softmax over 64 classes: one wave32 per row ----------------
__global__ void log_softmax64(const float* __restrict__ h, float* __restrict__ out, int M) {
    int wave = (blockIdx.x * blockDim.x + threadIdx.x) >> 5;
    int lane = threadIdx.x & 31;
    if (wave >= M) return;
    const float* row = h + (size_t)wave * 64;
    float v0 = row[lane];
    float v1 = row[lane + 32];
    float m = fmaxf(v0, v1);
#pragma unroll
    for (int o = 16; o > 0; o >>= 1) m = fmaxf(m, __shfl_xor(m, o, 32));
    float s = expf(v0 - m) + expf(v1 - m);
#pragma unroll
    for (int o = 16; o > 0; o >>= 1) s += __shfl_xor(s, o, 32);
    float lse = m + logf(s);
    float* orow = out + (size_t)wave * 64;
    orow[lane]      = v0 - lse;
    orow[lane + 32] = v1 - lse;
}

extern "C" void kernel_launch(void* const* d_in, const int* in_sizes, int n_in,
                              void* d_out, int out_size, void* d_ws, size_t ws_size,
                              hipStream_t stream) {
    (void)in_sizes; (void)n_in; (void)out_size; (void)ws_size;
    const float* x   = (const float*)d_in[0];
    const int*   ei  = (const int*)d_in[1];
    const float* Wl0 = (const float*)d_in[2];
    const float* Wr0 = (const float*)d_in[3];
    const float* b0  = (const float*)d_in[4];
    const float* Wl1 = (const float*)d_in[5];
    const float* Wr1 = (const float*)d_in[6];
    const float* b1  = (const float*)d_in[7];
    const float* Wl2 = (const float*)d_in[8];
    const float* Wr2 = (const float*)d_in[9];
    const float* b2  = (const float*)d_in[10];

    const int N = NN, E = NE;
    const int* src = ei;
    const int* dst = ei + E;

    // workspace carve (256B-aligned chunks)
    float* cnt = (float*)d_ws;                 // N floats (padded to 50048)
    float* agg = cnt + 50048;                  // N*128
    float* h1  = agg + (size_t)N * 128;        // N*128
    float* h2  = h1  + (size_t)N * 128;        // N*128

    float* outLS = (float*)d_out;              // [N,64] log_softmax
    float* h3    = outLS + (size_t)N * 64;     // [N,64] final pre-softmax h (2nd tuple elem)

    const int TPB = 256;
    const int nAgg = N * 128;
    const int pairsM = ((N / 16) + 1) / 2;     // 1563

    // degree counts (shared by all layers)
    zero_f32<<<(N + TPB - 1) / TPB, TPB, 0, stream>>>(cnt, N);
    degree_kernel<<<(E + TPB - 1) / TPB, TPB, 0, stream>>>(dst, cnt, E);

    // ---- layer 0: in x[ ,128] -> h1[ ,128], ReLU ----
    zero_f32<<<(nAgg + TPB - 1) / TPB, TPB, 0, stream>>>(agg, nAgg);
    scatter_add_kernel<<<((E << 5) + TPB - 1) / TPB, TPB, 0, stream>>>(x, src, dst, agg, E, 5);
    {
        int waves = pairsM * (128 / 16);
        sage_gemm_wmma<<<(waves + 7) / 8, TPB, 0, stream>>>(agg, cnt, x, Wl0, Wr0, b0, h1, N, 128, 128, 1);
    }

    // ---- layer 1: h1 -> h2, ReLU ----
    zero_f32<<<(nAgg + TPB - 1) / TPB, TPB, 0, stream>>>(agg, nAgg);
    scatter_add_kernel<<<((E << 5) + TPB - 1) / TPB, TPB, 0, stream>>>(h1, src, dst, agg, E, 5);
    {
        int waves = pairsM * (128 / 16);
        sage_gemm_wmma<<<(waves + 7) / 8, TPB, 0, stream>>>(agg, cnt, h1, Wl1, Wr1, b1, h2, N, 128, 128, 1);
    }

    // ---- layer 2: h2 -> h3[ ,64], no ReLU (written straight into d_out tail) ----
    zero_f32<<<(nAgg + TPB - 1) / TPB, TPB, 0, stream>>>(agg, nAgg);
    scatter_add_kernel<<<((E << 5) + TPB - 1) / TPB, TPB, 0, stream>>>(h2, src, dst, agg, E, 5);
    {
        int waves = pairsM * (64 / 16);
        sage_gemm_wmma<<<(waves + 7) / 8, TPB, 0, stream>>>(agg, cnt, h2, Wl2, Wr2, b2, h3, N, 128, 64, 0);
    }

    // ---- log_softmax over 64 classes ----
    log_softmax64<<<((N << 5) + TPB - 1) / TPB, TPB, 0, stream>>>(h3, outLS, N);
}